// LightGCNModel_80238579023928
// MI455X (gfx1250) — compile-verified
//
#include <hip/hip_runtime.h>

#define N_USERS 100000
#define N_ITEMS 50000
#define N_NODES 150000
#define NNZ     4800000
#define EMB     64

typedef __attribute__((ext_vector_type(2))) float v2f;
typedef __attribute__((ext_vector_type(8))) float v8f;

// ---------------------------------------------------------------------------
// init: acc = h = concat(user_emb, item_emb)   (float4 granularity)
// ---------------------------------------------------------------------------
__global__ void k_init_concat(const float4* __restrict__ u,
                              const float4* __restrict__ it,
                              float4* __restrict__ acc,
                              float4* __restrict__ h) {
    long i = (long)blockIdx.x * blockDim.x + threadIdx.x;     // float4 index
    const long nU = (long)N_USERS * (EMB / 4);
    const long nT = (long)N_NODES * (EMB / 4);
    if (i >= nT) return;
    float4 v = (i < nU) ? u[i] : it[i - nU];
    acc[i] = v;
    h[i]   = v;
}

__global__ void k_zero(float4* __restrict__ p, long n4) {
    long i = (long)blockIdx.x * blockDim.x + threadIdx.x;
    if (i < n4) p[i] = make_float4(0.f, 0.f, 0.f, 0.f);
}

// ---------------------------------------------------------------------------
// COO SpMM scatter: one wave32 per edge; each lane handles 2 dims (float2)
// Atomics land in L2 (38.4MB accumulator fits easily in 192MB L2).
// ---------------------------------------------------------------------------
__global__ __launch_bounds__(256)
void k_spmm_scatter(const int*   __restrict__ rows,
                    const int*   __restrict__ cols,
                    const float* __restrict__ vals,
                    const float* __restrict__ x,
                    float*       __restrict__ y) {
    long tid  = (long)blockIdx.x * blockDim.x + threadIdx.x;
    long edge = tid >> 5;
    int  lane = (int)(tid & 31);
    if (edge >= NNZ) return;
    int   r = rows[edge];
    int   c = cols[edge];
    float v = vals[edge];
    const float2* src = (const float2*)(x + (long)c * EMB);
    float2 g = src[lane];
    float* dst = y + (long)r * EMB + lane * 2;
    atomicAdd(dst + 0, g.x * v);
    atomicAdd(dst + 1, g.y * v);
}

__global__ void k_acc_add(float4* __restrict__ acc, const float4* __restrict__ h, long n4) {
    long i = (long)blockIdx.x * blockDim.x + threadIdx.x;
    if (i >= n4) return;
    float4 a = acc[i], b = h[i];
    a.x += b.x; a.y += b.y; a.z += b.z; a.w += b.w;
    acc[i] = a;
}

// ---------------------------------------------------------------------------
// finalize: combined = acc*0.25 (in place inside d_out = [user|item]);
// masked_user = combined*mask; mask copied to output slot 4.
// ---------------------------------------------------------------------------
__global__ void k_finalize(float4* __restrict__ comb,
                           const float4* __restrict__ mask,
                           float4* __restrict__ masked_out,
                           float4* __restrict__ mask_out) {
    long i = (long)blockIdx.x * blockDim.x + threadIdx.x;     // float4 index
    const long nU = (long)N_USERS * (EMB / 4);
    const long nT = (long)N_NODES * (EMB / 4);
    if (i >= nT) return;
    float4 c = comb[i];
    c.x *= 0.25f; c.y *= 0.25f; c.z *= 0.25f; c.w *= 0.25f;
    comb[i] = c;
    if (i < nU) {
        float4 m = mask[i];
        float4 mc = make_float4(c.x * m.x, c.y * m.y, c.z * m.z, c.w * m.w);
        masked_out[i] = mc;
        mask_out[i]   = m;
    }
}

// ---------------------------------------------------------------------------
// Tail GEMM with CDNA5 WMMA: out = relu(A[M,64] @ W[64,64] + bias)
// V_WMMA_F32_16X16X4_F32, one wave per 16-row tile, 4 N-tiles, K unrolled.
//
// A fragments: lane needs {A[m][4j+2hi], A[m][4j+2hi+1]} -> one b64 load.
//   OOB rows are address-clamped (their D rows are never stored), so the
//   loads are unconditional and EXEC stays all-ones for WMMA.
// B fragments: W repacked in LDS as v2f pairs {W[2p][n], W[2p+1][n]} so each
//   fragment is a single ds_load_b64 (no v_dual_mov assembly).
// ---------------------------------------------------------------------------
__global__ __launch_bounds__(256)
void k_gemm_wmma_f32(const float* __restrict__ A,
                     const float* __restrict__ W,
                     const float* __restrict__ bias,
                     float* __restrict__ out,
                     int M) {
    __shared__ v2f   sWp[32 * EMB];   // 16 KB: sWp[p*64 + n] = {W[2p][n], W[2p+1][n]}
    __shared__ float sB[EMB];

    for (int i = threadIdx.x; i < EMB * EMB; i += 256) {
        const int k = i >> 6;          // row of W
        const int n = i & 63;          // col of W
        ((float*)&sWp[(k >> 1) * EMB + n])[k & 1] = W[i];
    }
    if (threadIdx.x < EMB) sB[threadIdx.x] = bias[threadIdx.x];
    __syncthreads();

    const int waveId = threadIdx.x >> 5;
    const int lane   = threadIdx.x & 31;
    const int m16    = lane & 15;     // A row within tile / D column
    const int hi     = lane >> 4;     // K sub-pair select (A/B), row+8 select (D)

    const long rowBase = (long)blockIdx.x * 128 + (long)waveId * 16;
    const long aRow    = rowBase + m16;
    const long aSafe   = aRow < (long)M ? aRow : (long)(M - 1);   // clamp, no branch
    const v2f* __restrict__ a2 = (const v2f*)(A + aSafe * EMB);

    v8f c0 = {}, c1 = {}, c2 = {}, c3 = {};

#pragma unroll
    for (int j = 0; j < 16; ++j) {                 // kb = 4*j
        const v2f a = a2[2 * j + hi];              // global_load_b64, unconditional
        const v2f* __restrict__ wp = &sWp[(2 * j + hi) * EMB + m16];
        const v2f b0 = wp[0];
        const v2f b1 = wp[16];
        const v2f b2 = wp[32];
        const v2f b3 = wp[48];
        c0 = __builtin_amdgcn_wmma_f32_16x16x4_f32(false, a, false, b0, (short)0, c0, false, false);
        c1 = __builtin_amdgcn_wmma_f32_16x16x4_f32(false, a, false, b1, (short)0, c1, false, false);
        c2 = __builtin_amdgcn_wmma_f32_16x16x4_f32(false, a, false, b2, (short)0, c2, false, false);
        c3 = __builtin_amdgcn_wmma_f32_16x16x4_f32(false, a, false, b3, (short)0, c3, false, false);
    }

    // Per-column bias, hoisted out of the store loop
    const float bb0 = sB[ 0 + m16];
    const float bb1 = sB[16 + m16];
    const float bb2 = sB[32 + m16];
    const float bb3 = sB[48 + m16];

    // D layout: VGPR r -> row (r + 8*hi), column = m16 (+ n-tile offset)
#pragma unroll
    for (int r = 0; r < 8; ++r) {
        const long orow = rowBase + r + 8 * hi;
        if (orow < (long)M) {
            float* o = out + orow * EMB;
            float v;
            v = c0[r] + bb0; o[ 0 + m16] = v > 0.f ? v : 0.f;
            v = c1[r] + bb1; o[16 + m16] = v > 0.f ? v : 0.f;
            v = c2[r] + bb2; o[32 + m16] = v > 0.f ? v : 0.f;
            v = c3[r] + bb3; o[48 + m16] = v > 0.f ? v : 0.f;
        }
    }
}

// ---------------------------------------------------------------------------
extern "C" void kernel_launch(void* const* d_in, const int* in_sizes, int n_in,
                              void* d_out, int out_size, void* d_ws, size_t ws_size,
                              hipStream_t stream) {
    (void)in_sizes; (void)n_in; (void)out_size; (void)ws_size;

    const float* user_emb = (const float*)d_in[0];
    const float* item_emb = (const float*)d_in[1];
    const int*   adj_rows = (const int*)  d_in[2];
    const int*   adj_cols = (const int*)  d_in[3];
    const float* adj_vals = (const float*)d_in[4];
    const float* mask     = (const float*)d_in[5];
    const float* attr_W   = (const float*)d_in[6];
    const float* attr_b   = (const float*)d_in[7];

    float* out = (float*)d_out;
    float* out_comb   = out;                                   // [user|item] = combined
    float* out_masked = out + (long)N_NODES * EMB;
    float* out_pred   = out_masked + (long)N_USERS * EMB;
    float* out_maskcp = out_pred   + (long)N_USERS * EMB;

    float* hA = (float*)d_ws;                                  // 38.4 MB
    float* hB = hA + (long)N_NODES * EMB;                      // 38.4 MB

    const long n4    = (long)N_NODES * (EMB / 4);              // 2.4M float4
    const int  elemB = (int)((n4 + 255) / 256);
    const int  spmmB = (NNZ * 32 + 255) / 256;                 // wave per edge

    // acc (in d_out) = h (hA) = concat(user, item)
    k_init_concat<<<elemB, 256, 0, stream>>>((const float4*)user_emb,
                                             (const float4*)item_emb,
                                             (float4*)out_comb, (float4*)hA);

    // 3 propagation layers, ping-pong hA/hB, acc += h each layer
    float* cur = hA; float* nxt = hB;
    for (int layer = 0; layer < 3; ++layer) {
        k_zero<<<elemB, 256, 0, stream>>>((float4*)nxt, n4);
        k_spmm_scatter<<<spmmB, 256, 0, stream>>>(adj_rows, adj_cols, adj_vals, cur, nxt);
        k_acc_add<<<elemB, 256, 0, stream>>>((float4*)out_comb, (const float4*)nxt, n4);
        float* t = cur; cur = nxt; nxt = t;
    }

    // combined = acc/4 (in place), masked user emb, mask passthrough
    k_finalize<<<elemB, 256, 0, stream>>>((float4*)out_comb, (const float4*)mask,
                                          (float4*)out_masked, (float4*)out_maskcp);

    // predicted_attributes = relu(masked @ W + b) via V_WMMA_F32_16X16X4_F32
    const int gemmB = (N_USERS + 127) / 128;                   // 128 rows per block
    k_gemm_wmma_f32<<<gemmB, 256, 0, stream>>>(out_masked, attr_W, attr_b,
                                               out_pred, N_USERS);
}